// HPO_Bbox_AR_Loss_85074712199850
// MI455X (gfx1250) — compile-verified
//
#include <hip/hip_runtime.h>
#include <hip/hip_bf16.h>
#include <math.h>

// ---------------- constants ----------------
#define NA 5
#define NCH 42          // 5 + 13 + 24 channels per anchor
#define GRID 52
#define PLANE (GRID*GRID)       // 2704
#define BATCH 64
#define NUM_ACTION 13
#define NUM_OBJ 24
#define SIL_THRESH 0.6f
#define OBJECT_SCALE 5.0f

// dense kernel geometry: 64*5*52*52 cells = 865,280; 4 cells/thread -> 216,320 threads
#define CELLS_PER_THREAD 4
#define BLOCK_SZ 256
#define NBLOCKS_MAIN 845        // 216320 / 256 exactly

__constant__ float c_anchors[2*NA] = {
    1.3221f, 1.73145f,
    3.19275f, 4.00944f,
    5.05587f, 8.09892f,
    9.47112f, 4.84053f,
    11.2364f, 10.0071f
};

typedef __attribute__((ext_vector_type(2))) float v2f;
typedef __attribute__((ext_vector_type(4))) float v4f;
typedef __attribute__((ext_vector_type(8))) float v8f;

__device__ __forceinline__ float sigm(float x) {
    return 1.0f / (1.0f + __expf(-x));
}

// Wave-level reduction using V_WMMA_F32_16X16X4_F32.
// A (16x4 f32): lanes 0-15 supply K=0 (VGPR0) / K=1 (VGPR1); lanes 16-31 K=2/K=3.
// Put the lane partial in K-even slot, 0 in K-odd; B = all-ones 4x16.
// D[m][n] = v(lane m) + v(lane m+16) for every n, so summing the 8 D VGPRs gives
// the half-wave sums; one xor-16 shuffle produces the full wave sum in all lanes.
__device__ __forceinline__ float wave_reduce_wmma(float v) {
    v2f a; a.x = v; a.y = 0.0f;
    v2f b; b.x = 1.0f; b.y = 1.0f;
    v8f c = {};
    v8f d = __builtin_amdgcn_wmma_f32_16x16x4_f32(
        /*neg_a=*/false, a, /*neg_b=*/false, b,
        /*c_mod=*/(short)0, c, /*reuse_a=*/false, /*reuse_b=*/false);
    float s = d[0] + d[1] + d[2] + d[3] + d[4] + d[5] + d[6] + d[7];
    s += __shfl_xor(s, 16, 32);
    return s;
}

// ---------------------------------------------------------------------------
// Kernel A: per-batch sparse work (64 threads, 1 block).
// Computes the correction term (responsible cell: swap default targets for the
// real ones, override conf mask with OBJECT_SCALE/iou_t) + both cross-entropies.
// ---------------------------------------------------------------------------
__global__ __launch_bounds__(64)
void batch_terms_kernel(const float* __restrict__ pred,
                        const float* __restrict__ bbox,
                        const int*   __restrict__ act_gt,
                        const int*   __restrict__ obj_gt,
                        float*       __restrict__ ws_batch) {
    const int b = threadIdx.x;

    const float gx = bbox[b*4 + 0] * (float)GRID;
    const float gy = bbox[b*4 + 1] * (float)GRID;
    const float gw = bbox[b*4 + 2] * (float)GRID;
    const float gh = bbox[b*4 + 3] * (float)GRID;
    const int gi = (int)gx;
    const int gj = (int)gy;

    // best anchor: IoU of centered boxes = min(w)*min(h) / (areas - inter); first max wins
    float best_iou = -1.0f; int best = 0;
    #pragma unroll
    for (int k = 0; k < NA; ++k) {
        const float aw = c_anchors[2*k], ah = c_anchors[2*k + 1];
        const float inter = fminf(gw, aw) * fminf(gh, ah);
        const float iou = inter / (gw*gh + aw*ah - inter);
        if (iou > best_iou) { best_iou = iou; best = k; }
    }
    const float aw = c_anchors[2*best], ah = c_anchors[2*best + 1];

    const int base = ((b*(NA*NCH) + best*NCH)*GRID + gj)*GRID + gi;

    const float t0 = pred[base + 0*PLANE];
    const float t1 = pred[base + 1*PLANE];
    const float t2 = pred[base + 2*PLANE];
    const float t3 = pred[base + 3*PLANE];
    const float t4 = pred[base + 4*PLANE];

    const float tx = sigm(t0), ty = sigm(t1), conf = sigm(t4);
    const float px = tx + (float)gi;
    const float py = ty + (float)gj;
    const float pw = __expf(t2) * aw;
    const float ph = __expf(t3) * ah;

    // iou_t of GT box vs predicted box at the responsible cell
    const float mx = fminf(gx - gw*0.5f, px - pw*0.5f);
    const float Mx = fmaxf(gx + gw*0.5f, px + pw*0.5f);
    const float my = fminf(gy - gh*0.5f, py - ph*0.5f);
    const float My = fmaxf(gy + gh*0.5f, py + ph*0.5f);
    const float cw = gw + pw - (Mx - mx);
    const float ch = gh + ph - (My - my);
    const float inter = (cw > 0.0f && ch > 0.0f) ? cw*ch : 0.0f;
    const float iou_t = inter / (gw*gh + pw*ph - inter);
    const float iou_cell = fmaxf(iou_t, 0.0f);
    const float mask2_cell = (iou_cell > SIL_THRESH) ? 0.0f : 1.0f;

    // real targets at this cell
    const float tbx = gx - (float)gi;
    const float tby = gy - (float)gj;
    const float tbw = __logf(gw / aw);
    const float tbh = __logf(gh / ah);

    // correction = (real contribution) - (default contribution already in dense sum)
    const float dx  = tx - tbx,  dx0 = tx - 0.5f;
    const float dy  = ty - tby,  dy0 = ty - 0.5f;
    const float dw  = t2 - tbw;
    const float dh  = t3 - tbh;
    const float dc  = conf - iou_t;
    float corr = 0.5f * ( dx*dx - dx0*dx0
                        + dy*dy - dy0*dy0
                        + dw*dw - t2*t2
                        + dh*dh - t3*t3
                        + OBJECT_SCALE*dc*dc - mask2_cell*conf*conf );

    // cross entropies over action (13) and object (24) logits at this cell
    const int albl = act_gt[b];
    const int olbl = obj_gt[b];
    const int abase = base + 5*PLANE;
    const int obase = base + (5 + NUM_ACTION)*PLANE;

    float am = -1e30f, apick = 0.0f;
    #pragma unroll
    for (int k = 0; k < NUM_ACTION; ++k) {
        const float v = pred[abase + k*PLANE];
        am = fmaxf(am, v);
        if (k == albl) apick = v;
    }
    float as = 0.0f;
    #pragma unroll
    for (int k = 0; k < NUM_ACTION; ++k)
        as += __expf(pred[abase + k*PLANE] - am);
    const float loss_act = (am + __logf(as)) - apick;

    float om = -1e30f, opick = 0.0f;
    #pragma unroll
    for (int k = 0; k < NUM_OBJ; ++k) {
        const float v = pred[obase + k*PLANE];
        om = fmaxf(om, v);
        if (k == olbl) opick = v;
    }
    float os = 0.0f;
    #pragma unroll
    for (int k = 0; k < NUM_OBJ; ++k)
        os += __expf(pred[obase + k*PLANE] - om);
    const float loss_obj = (om + __logf(os)) - opick;

    ws_batch[b] = corr + loss_act + loss_obj;
}

// ---------------------------------------------------------------------------
// Kernel B: dense streaming part. Reads ONLY channels 0..4 of each anchor
// (17.3 MB of the 145 MB tensor). Each thread: 4 consecutive cells along x,
// 5 coalesced float4 loads. No divergence -> full-EXEC WMMA wave reduction.
// ---------------------------------------------------------------------------
__global__ __launch_bounds__(BLOCK_SZ)
void dense_loss_kernel(const float* __restrict__ pred,
                       const float* __restrict__ bbox,
                       float*       __restrict__ ws_partials) {
    const int idx = blockIdx.x * BLOCK_SZ + threadIdx.x;  // [0, 216320)
    int r = idx;
    const int i4 = r % (PLANE/4); r /= (PLANE/4);         // float4 index along row-major plane
    const int j  = r % GRID;      r /= GRID;
    const int a  = r % NA;
    const int b  = r / NA;

    const float gx = bbox[b*4 + 0] * (float)GRID;
    const float gy = bbox[b*4 + 1] * (float)GRID;
    const float gw = bbox[b*4 + 2] * (float)GRID;
    const float gh = bbox[b*4 + 3] * (float)GRID;
    const float aw = c_anchors[2*a], ah = c_anchors[2*a + 1];

    const int base = ((b*(NA*NCH) + a*NCH)*GRID + j)*GRID + i4*4;

    const v4f v0 = *reinterpret_cast<const v4f*>(pred + base + 0*PLANE);
    const v4f v1 = *reinterpret_cast<const v4f*>(pred + base + 1*PLANE);
    const v4f v2 = *reinterpret_cast<const v4f*>(pred + base + 2*PLANE);
    const v4f v3 = *reinterpret_cast<const v4f*>(pred + base + 3*PLANE);
    const v4f v4 = *reinterpret_cast<const v4f*>(pred + base + 4*PLANE);

    const float jf = (float)j;
    const float gxl = gx - gw*0.5f, gxh = gx + gw*0.5f;
    const float gyl = gy - gh*0.5f, gyh = gy + gh*0.5f;
    const float garea = gw * gh;

    float local = 0.0f;
    // Column i4*4+e (plane is row-major, row stride GRID=52 which is 4-aligned).
    #pragma unroll
    for (int e = 0; e < 4; ++e) {
        const float tx   = sigm(v0[e]);
        const float ty   = sigm(v1[e]);
        const float tw   = v2[e];
        const float th   = v3[e];
        const float conf = sigm(v4[e]);

        const float px = tx + (float)(i4*4 + e);
        const float py = ty + jf;
        const float pw = __expf(tw) * aw;
        const float ph = __expf(th) * ah;

        const float mx = fminf(gxl, px - pw*0.5f);
        const float Mx = fmaxf(gxh, px + pw*0.5f);
        const float my = fminf(gyl, py - ph*0.5f);
        const float My = fmaxf(gyh, py + ph*0.5f);
        const float cw = gw + pw - (Mx - mx);
        const float ch = gh + ph - (My - my);
        const float inter = (cw > 0.0f && ch > 0.0f) ? cw*ch : 0.0f;
        const float iou = fmaxf(inter / (garea + pw*ph - inter), 0.0f);
        const float mask2 = (iou > SIL_THRESH) ? 0.0f : 1.0f;   // NO_OBJECT_SCALE = 1

        const float ex = tx - 0.5f;
        const float ey = ty - 0.5f;
        local += 0.5f * (ex*ex + ey*ey + tw*tw + th*th + mask2*conf*conf);
    }

    // block reduction: WMMA wave reduce -> 8 wave sums in LDS -> thread 0
    const float wsum = wave_reduce_wmma(local);
    __shared__ float sm[BLOCK_SZ/32];
    const int lane = threadIdx.x & 31;
    const int wid  = threadIdx.x >> 5;
    if (lane == 0) sm[wid] = wsum;
    __syncthreads();
    if (threadIdx.x == 0) {
        float t = 0.0f;
        #pragma unroll
        for (int k = 0; k < BLOCK_SZ/32; ++k) t += sm[k];
        ws_partials[blockIdx.x] = t;
    }
}

// ---------------------------------------------------------------------------
// Kernel C: deterministic final sum (fixed thread->index assignment, fixed
// tree order). Double accumulation for a ~1e6-magnitude sum of ~900 partials.
// ---------------------------------------------------------------------------
__global__ __launch_bounds__(BLOCK_SZ)
void final_reduce_kernel(const float* __restrict__ ws, int n,
                         float* __restrict__ out) {
    __shared__ double sm[BLOCK_SZ];
    double s = 0.0;
    for (int k = threadIdx.x; k < n; k += BLOCK_SZ) s += (double)ws[k];
    sm[threadIdx.x] = s;
    __syncthreads();
    for (int off = BLOCK_SZ/2; off > 0; off >>= 1) {
        if (threadIdx.x < off) sm[threadIdx.x] += sm[threadIdx.x + off];
        __syncthreads();
    }
    if (threadIdx.x == 0) out[0] = (float)sm[0];
}

// ---------------------------------------------------------------------------
extern "C" void kernel_launch(void* const* d_in, const int* in_sizes, int n_in,
                              void* d_out, int out_size, void* d_ws, size_t ws_size,
                              hipStream_t stream) {
    const float* pred = (const float*)d_in[0];   // (64, 210, 52, 52) f32
    const float* bbox = (const float*)d_in[1];   // (64, 4) f32
    const int*   act  = (const int*)d_in[2];     // (64,) i32
    const int*   obj  = (const int*)d_in[3];     // (64,) i32
    float* out = (float*)d_out;
    float* wsf = (float*)d_ws;
    // ws layout: [0,64) per-batch terms; [64, 64+NBLOCKS_MAIN) block partials
    (void)in_sizes; (void)n_in; (void)out_size; (void)ws_size;

    batch_terms_kernel<<<1, 64, 0, stream>>>(pred, bbox, act, obj, wsf);
    dense_loss_kernel<<<NBLOCKS_MAIN, BLOCK_SZ, 0, stream>>>(pred, bbox, wsf + BATCH);
    final_reduce_kernel<<<1, BLOCK_SZ, 0, stream>>>(wsf, BATCH + NBLOCKS_MAIN, out);
}